// ATAssigner_75471165325890
// MI455X (gfx1250) — compile-verified
//
#include <hip/hip_runtime.h>
#include <hip/hip_bf16.h>

#define NUM_CLASSES 80
#define BG_IDX 80
#define TOPK 9
#define NCAND (3 * TOPK)
#define BS 32
#define NMAX 64
#define NA_TOT 8400

typedef float v2f __attribute__((ext_vector_type(2)));
typedef float v8f __attribute__((ext_vector_type(8)));

__device__ __forceinline__ float iou_box(float4 g, float4 a) {
    float ix = fminf(g.z, a.z) - fmaxf(g.x, a.x);
    float iy = fminf(g.w, a.w) - fmaxf(g.y, a.y);
    ix = fmaxf(ix, 0.0f);
    iy = fmaxf(iy, 0.0f);
    float inter = ix * iy;
    float ag = (g.z - g.x) * (g.w - g.y);
    float aa = (a.z - a.x) * (a.w - a.y);
    return inter / (ag + aa - inter + 1e-9f);
}

// Monotonic sort key: dist^2 >= 0, so IEEE bits compare correctly as unsigned.
// Low 32 bits carry the anchor index -> ties resolve to the LOWER index,
// exactly matching jax.lax.top_k's stable ordering. Single v_cmp_lt_u64.
__device__ __forceinline__ unsigned long long dist_key(float d, int idx) {
    return ((unsigned long long)__float_as_uint(d) << 32) | (unsigned int)idx;
}

// One block per batch image. Wave w owns GTs [16w, 16w+16). Each wave sweeps
// anchor tiles of 16; V_WMMA_F32_16X16X4_F32 produces a 16x16 tile of squared
// center distances (K=4 encodes [x,y,1,||c||^2] x [-2x',-2y',||c'||^2,1]).
// D layout: lane L holds column (GT) L&15; VGPR r holds row (anchor) r (+8 for
// lanes>=16). Per-lane branchless top-9 over its 8-row half; halves merged in LDS.
__global__ __launch_bounds__(128) void atss_topk_kernel(
    const float* __restrict__ anchors, const float* __restrict__ gt_boxes,
    const float* __restrict__ mask_gt, unsigned long long* __restrict__ mask_pos) {
    const int b = blockIdx.x;
    const int tid = threadIdx.x;
    const int w = tid >> 5;
    const int lane = tid & 31;
    const int col = lane & 15;
    const int hi = (lane >> 4) & 1;
    const int m = w * 16 + col;  // this lane's GT column

    const float4* __restrict__ anc4 = (const float4*)anchors;
    const float4 g = ((const float4*)gt_boxes)[b * NMAX + m];
    const float gcx = 0.5f * (g.x + g.z), gcy = 0.5f * (g.y + g.w);

    // B operand (4x16): lanes 0-15 supply K=0,1; lanes 16-31 supply K=2,3.
    v2f Bop;
    Bop.x = hi ? (gcx * gcx + gcy * gcy) : (-2.0f * gcx);
    Bop.y = hi ? 1.0f : (-2.0f * gcy);

    __shared__ unsigned long long sk[4][32][TOPK];  // per-lane sorted key lists
    __shared__ int cidx[4][16][NCAND];              // merged candidate indices

    int start = 0;
    const int nas[3] = {6400, 1600, 400};
    for (int lvl = 0; lvl < 3; ++lvl) {
        const int na = nas[lvl];
        unsigned long long tk[TOPK];
#pragma unroll
        for (int j = 0; j < TOPK; ++j) tk[j] = ~0ull;

        for (int n0 = start; n0 < start + na; n0 += 16) {
            // A operand (16x4): row = anchor n0+col; lanes 0-15 give K=0,1,
            // lanes 16-31 give K=2,3 of the SAME row.
            const float4 ab = anc4[n0 + col];
            __builtin_prefetch(anc4 + n0 + 16 + col, 0, 0);  // global_prefetch_b8
            const float acx = 0.5f * (ab.x + ab.z), acy = 0.5f * (ab.y + ab.w);
            v2f Aop;
            Aop.x = hi ? 1.0f : acx;
            Aop.y = hi ? (acx * acx + acy * acy) : acy;
            v8f d;
#if defined(__gfx1250__) && __has_builtin(__builtin_amdgcn_wmma_f32_16x16x4_f32)
            v8f c = {};
            d = __builtin_amdgcn_wmma_f32_16x16x4_f32(
                /*neg_a=*/false, Aop, /*neg_b=*/false, Bop,
                /*c_mod=*/(short)0, c, /*reuse_a=*/false, /*reuse_b=*/false);
#else
            // Semantics-equivalent fallback via cross-lane shuffles.
#pragma unroll
            for (int r = 0; r < 8; ++r) {
                int p = r + hi * 8;
                float rax = __shfl(acx, p, 32);
                float ray = __shfl(acy, p, 32);
                float dx = rax - gcx, dy = ray - gcy;
                d[r] = dx * dx + dy * dy;
            }
#endif
            // Clamp fp32 cancellation (-eps) so bit-ordering stays monotonic.
            float dmin = 3.0e38f;
#pragma unroll
            for (int r = 0; r < 8; ++r) {
                d[r] = fmaxf(d[r], 0.0f);
                dmin = fminf(dmin, d[r]);
            }
            // Wave-uniform skip: run the insertion network only if some lane's
            // best value this tile could enter its top-9.
            if (__any(dist_key(dmin, 0) < tk[TOPK - 1])) {
                const int rowbase = n0 + hi * 8;
#pragma unroll
                for (int r = 0; r < 8; ++r) {
                    unsigned long long kk = dist_key(d[r], rowbase + r);
#pragma unroll
                    for (int j = 0; j < TOPK; ++j) {  // branchless sorted insert
                        unsigned long long o = tk[j];
                        bool take = kk < o;
                        tk[j] = take ? kk : o;
                        kk = take ? o : kk;
                    }
                }
            }
        }
#pragma unroll
        for (int j = 0; j < TOPK; ++j) sk[w][lane][j] = tk[j];
        __syncthreads();
        if (!hi) {  // two-pointer merge of the two sorted half-lists per GT
            int p0 = 0, p1 = 0;
#pragma unroll
            for (int j = 0; j < TOPK; ++j) {
                unsigned long long k0 = sk[w][col][p0];
                unsigned long long k1 = sk[w][col + 16][p1];
                bool t0 = k0 < k1;
                cidx[w][col][lvl * TOPK + j] = (int)(unsigned int)(t0 ? k0 : k1);
                p0 += t0 ? 1 : 0;
                p1 += t0 ? 0 : 1;
            }
        }
        __syncthreads();
        start += na;
    }

    // Phase 2 (lanes 0-15, one GT each): gather candidate IoUs, adaptive
    // threshold (mean + sample std), center-inside test, set mask_pos bits.
    if (!hi) {
        float civ[NCAND];
        int cin[NCAND];
        float s = 0.f, s2 = 0.f;
#pragma unroll
        for (int j = 0; j < NCAND; ++j) {
            int ai = cidx[w][col][j];
            cin[j] = ai;
            float v = iou_box(g, anc4[ai]);
            civ[j] = v;
            s += v;
            s2 += v * v;
        }
        const float k = (float)NCAND;
        float mean = s / k;
        float var = (s2 - k * mean * mean) / (k - 1.0f);
        float thr = mean + sqrtf(fmaxf(var, 0.0f));
        float mg = mask_gt[b * NMAX + m];
        if (mg > 0.0f) {
            const unsigned long long bit = 1ull << m;
#pragma unroll
            for (int j = 0; j < NCAND; ++j) {
                int ai = cin[j];
                float4 ab = anc4[ai];
                float acx = 0.5f * (ab.x + ab.z), acy = 0.5f * (ab.y + ab.w);
                float dlt = fminf(acx - g.x, acy - g.y);
                float drb = fminf(g.z - acx, g.w - acy);
                bool inside = fminf(dlt, drb) > 1e-9f;
                if (civ[j] > thr && inside) {
                    atomicOr(&mask_pos[(size_t)b * NA_TOT + ai], bit);
                }
            }
        }
    }
}

// Per-anchor finalization: resolve multi-GT anchors by branchless IoU argmax,
// gather targets, pred-IoU max over assigned GTs, write all four outputs.
__global__ __launch_bounds__(256) void atss_finalize_kernel(
    const float* __restrict__ anchors, const float* __restrict__ gt_boxes,
    const int* __restrict__ labels, const float* __restrict__ pred,
    const unsigned long long* __restrict__ mask_pos, float* __restrict__ out) {
    const int b = blockIdx.y;
    __shared__ float4 sg[NMAX];
    if (threadIdx.x < NMAX) sg[threadIdx.x] = ((const float4*)gt_boxes)[b * NMAX + threadIdx.x];
    __syncthreads();
    const int a = blockIdx.x * blockDim.x + threadIdx.x;
    if (a >= NA_TOT) return;

    unsigned long long mk = mask_pos[(size_t)b * NA_TOT + a];
    const float4 ab = ((const float4*)anchors)[a];
    if (__popcll(mk) > 1) {  // anchor claimed by >1 GT: keep argmax-IoU GT
        // key = (iou_bits << 32) | (NMAX-1-m): max-reduce, tie -> lowest m.
        unsigned long long best = 0ull;
#pragma unroll 8
        for (int mi = 0; mi < NMAX; ++mi) {
            float v = iou_box(sg[mi], ab);
            unsigned long long kk =
                ((unsigned long long)__float_as_uint(v) << 32) | (unsigned int)(NMAX - 1 - mi);
            best = (kk > best) ? kk : best;
        }
        mk = 1ull << (NMAX - 1 - (int)(unsigned int)best);
    }
    const bool fgb = (mk != 0ull);
    const int tgt = fgb ? (int)__builtin_ctzll(mk) : 0;
    const int lbl = fgb ? labels[b * NMAX + tgt] : BG_IDX;

    const float4 pb = ((const float4*)pred)[(size_t)b * NA_TOT + a];
    float ioumax = 0.0f;
    unsigned long long t = mk;
    while (t) {
        int mi = (int)__builtin_ctzll(t);
        t &= t - 1ull;
        ioumax = fmaxf(ioumax, iou_box(sg[mi], pb));
    }

    const long long o_box = (long long)BS * NA_TOT;
    const long long o_sco = o_box + (long long)BS * NA_TOT * 4;
    const long long o_fg  = o_sco + (long long)BS * NA_TOT * NUM_CLASSES;
    const long long base = (long long)b * NA_TOT + a;

    out[base] = (float)lbl;                       // t_labels (as float)
    const float4 tb = sg[tgt];
    ((float4*)(out + o_box))[base] = tb;          // t_boxes
    float4* so4 = (float4*)(out + o_sco + base * (long long)NUM_CLASSES);
#pragma unroll
    for (int c4 = 0; c4 < NUM_CLASSES / 4; ++c4) {  // t_scores: one-hot * ioumax
        float4 vv;
        vv.x = (4 * c4 + 0 == lbl) ? ioumax : 0.0f;
        vv.y = (4 * c4 + 1 == lbl) ? ioumax : 0.0f;
        vv.z = (4 * c4 + 2 == lbl) ? ioumax : 0.0f;
        vv.w = (4 * c4 + 3 == lbl) ? ioumax : 0.0f;
        so4[c4] = vv;
    }
    out[o_fg + base] = fgb ? 1.0f : 0.0f;         // fg_mask
}

extern "C" void kernel_launch(void* const* d_in, const int* in_sizes, int n_in,
                              void* d_out, int out_size, void* d_ws, size_t ws_size,
                              hipStream_t stream) {
    const float* anchors = (const float*)d_in[0];
    // d_in[1] = num_anchors_list (static [6400,1600,400], baked in)
    const int* labels = (const int*)d_in[2];
    const float* gtb = (const float*)d_in[3];
    const float* maskgt = (const float*)d_in[4];
    const float* pred = (const float*)d_in[5];

    unsigned long long* mp = (unsigned long long*)d_ws;  // 32*8400*8 = 2.15 MB
    hipMemsetAsync(mp, 0, (size_t)BS * NA_TOT * sizeof(unsigned long long), stream);

    atss_topk_kernel<<<BS, 128, 0, stream>>>(anchors, gtb, maskgt, mp);

    dim3 grid((NA_TOT + 255) / 256, BS);
    atss_finalize_kernel<<<grid, 256, 0, stream>>>(anchors, gtb, labels, pred, mp,
                                                   (float*)d_out);
}